// DynViT_47425028882619
// MI455X (gfx1250) — compile-verified
//
#include <hip/hip_runtime.h>

#define B_ 4
#define N_ 2048
#define C_ 768
#define K_ 16
#define H_ 512
#define W_ 512

typedef __attribute__((ext_vector_type(2))) float v2f;
typedef __attribute__((ext_vector_type(8))) float v8f;
typedef __attribute__((ext_vector_type(4))) int v4i;

#define GAS128(p) ((__attribute__((address_space(1))) v4i*)(const_cast<float*>(p)))
#define LAS128(p) ((__attribute__((address_space(3))) v4i*)(p))

#if __has_builtin(__builtin_amdgcn_global_load_async_to_lds_b128)
#define HAVE_ASYNC_LDS 1
#else
#define HAVE_ASYNC_LDS 0
#endif

__device__ __forceinline__ void wait_async0() {
#if __has_builtin(__builtin_amdgcn_s_wait_asynccnt)
  __builtin_amdgcn_s_wait_asynccnt(0);
#else
  asm volatile("s_wait_asynccnt 0" ::: "memory");
#endif
}

__device__ __forceinline__ float waveRedSum(float v) {
  for (int m = 16; m >= 1; m >>= 1) v += __shfl_xor(v, m, 32);
  return v;
}
__device__ __forceinline__ int waveRedMin(int v) {
  for (int m = 16; m >= 1; m >>= 1) v = min(v, __shfl_xor(v, m, 32));
  return v;
}

// ---------------- prototype / x row norms, cosine sims ----------------
__global__ void k_pnorm(const float* __restrict__ p, float* __restrict__ pninv) {
  int k = threadIdx.x;
  if (k >= K_) return;
  const float4* row = (const float4*)(p + (size_t)k * C_);
  float s = 0.f;
  for (int i = 0; i < C_ / 4; ++i) {
    float4 q = row[i];
    s += q.x * q.x + q.y * q.y + q.z * q.z + q.w * q.w;
  }
  pninv[k] = rsqrtf(s);
}

__global__ void k_xnorm(const float* __restrict__ x, float* __restrict__ xninv) {
  int i = blockIdx.x * blockDim.x + threadIdx.x;
  if (i >= B_ * N_) return;
  const float4* row = (const float4*)(x + (size_t)i * C_);
  float s = 0.f;
  for (int j = 0; j < C_ / 4; ++j) {
    float4 q = row[j];
    s += q.x * q.x + q.y * q.y + q.z * q.z + q.w * q.w;
  }
  xninv[i] = rsqrtf(s);
}

__global__ void k_cs(const float* __restrict__ x, const float* __restrict__ p,
                     const float* __restrict__ xninv, const float* __restrict__ pninv,
                     float* __restrict__ cs) {
  int i = blockIdx.x * blockDim.x + threadIdx.x;
  if (i >= B_ * N_ * K_) return;
  int k = i % K_;
  int bn = i / K_;
  const float4* xr = (const float4*)(x + (size_t)bn * C_);
  const float4* pr = (const float4*)(p + (size_t)k * C_);
  float s = 0.f;
  for (int j = 0; j < C_ / 4; ++j) {
    float4 a = xr[j], b = pr[j];
    s += a.x * b.x + a.y * b.y + a.z * b.z + a.w * b.w;
  }
  float v = (s * xninv[bn] * pninv[k] + 1.0f) * 0.5f;
  cs[(size_t)bn * K_ + k] = (v < 0.5f) ? 0.0f : v;
}

// ---------------- class_adj = cs * cs^T  via fp32 WMMA ----------------
__global__ void k_class_adj(const float* __restrict__ cs, float* __restrict__ adj) {
  const int TN = N_ / 16;  // 128
  int wave = (blockIdx.x * blockDim.x + threadIdx.x) >> 5;
  int lane = threadIdx.x & 31;
  if (wave >= B_ * TN * TN) return;
  int mt = wave % TN;
  int nt = (wave / TN) % TN;
  int b = wave / (TN * TN);
  int half = lane >> 4, l16 = lane & 15;
  const float* csb = cs + (size_t)b * N_ * K_;
  v8f acc = {};
  for (int kk = 0; kk < 4; ++kk) {
    int kb = kk * 4 + half * 2;
    v2f a = *(const v2f*)(csb + (size_t)(nt * 16 + l16) * K_ + kb);   // A[n][k..k+1]
    v2f bb = *(const v2f*)(csb + (size_t)(mt * 16 + l16) * K_ + kb);  // B[k..k+1][m] = cs[m][k..]
    acc = __builtin_amdgcn_wmma_f32_16x16x4_f32(false, a, false, bb, (short)0, acc, false, false);
  }
  float* out = adj + (size_t)b * N_ * N_;
  for (int v = 0; v < 8; ++v) {
    int row = nt * 16 + v + 8 * half;
    out[(size_t)row * N_ + mt * 16 + l16] = acc[v];
  }
}

// ---------------- proximity adjacency scatter ----------------
__global__ void k_prox(const int* __restrict__ seg, float* __restrict__ adjP) {
  long long i = (long long)blockIdx.x * blockDim.x + threadIdx.x;
  if (i >= (long long)B_ * H_ * W_) return;
  int b = (int)(i / (H_ * W_));
  int p = (int)(i % (H_ * W_));
  int y = p / W_, xq = p % W_;
  const int* s = seg + (size_t)b * H_ * W_;
  float* A = adjP + (size_t)b * N_ * N_;
  int a = s[p];
  if (xq + 1 < W_) {
    int c = s[p + 1];
    atomicAdd(&A[(size_t)a * N_ + c], 1.f);
    atomicAdd(&A[(size_t)c * N_ + a], 1.f);
  }
  if (y + 1 < H_) {
    int c = s[p + W_];
    atomicAdd(&A[(size_t)a * N_ + c], 1.f);
    atomicAdd(&A[(size_t)c * N_ + a], 1.f);
  }
}

// ---------------- symmetric-normalization degree^-1/2 ----------------
__global__ void k_dinv(const float* __restrict__ M, float* __restrict__ dinv) {
  int wave = (blockIdx.x * blockDim.x + threadIdx.x) >> 5;
  int lane = threadIdx.x & 31;
  if (wave >= B_ * N_) return;
  const float* row = M + (size_t)wave * N_;
  float s = 0.f;
  for (int m = lane; m < N_; m += 32) s += row[m];
  s = waveRedSum(s);
  if (lane == 0) dinv[wave] = (s > 0.f) ? rsqrtf(s) : 0.f;
}

// ---------------- adj = norm(prox) * norm(class), diag=1 ----------------
__global__ void k_fuse(const float* __restrict__ adjP, const float* __restrict__ dp,
                       const float* __restrict__ dc, float* __restrict__ adjW) {
  long long i = (long long)blockIdx.x * blockDim.x + threadIdx.x;
  if (i >= (long long)B_ * N_ * N_) return;
  int m = (int)(i % N_);
  long long t = i / N_;
  int n = (int)(t % N_);
  int b = (int)(t / N_);
  float v;
  if (n == m)
    v = 1.0f;
  else
    v = adjP[i] * dp[b * N_ + n] * dp[b * N_ + m] * adjW[i] * dc[b * N_ + n] * dc[b * N_ + m];
  adjW[i] = v;
}

// ---------------- connected components: min-label + pointer jump ----------------
__global__ void k_lab_init(int* lab) {
  int i = blockIdx.x * blockDim.x + threadIdx.x;
  if (i < B_ * N_) lab[i] = i % N_;
}

__global__ void k_cc(const float* __restrict__ adj, const int* __restrict__ lin,
                     int* __restrict__ lout) {
  int wave = (blockIdx.x * blockDim.x + threadIdx.x) >> 5;
  int lane = threadIdx.x & 31;
  if (wave >= B_ * N_) return;
  int b = wave / N_;
  const float* row = adj + (size_t)wave * N_;
  const int* lb = lin + b * N_;
  int v = 0x7fffffff;
  for (int m = lane; m < N_; m += 32)
    if (row[m] > 0.f) v = min(v, lb[m]);
  v = waveRedMin(v);
  if (lane == 0) {
    v = min(v, lb[wave % N_]);
    lout[wave] = lb[v];  // pointer jump: label of min neighbor's root
  }
}

// ---------------- label compaction + segment remap ----------------
__global__ void k_nmax(const int* __restrict__ seg, int* __restrict__ nmax) {
  long long i = (long long)blockIdx.x * blockDim.x + threadIdx.x;
  if (i >= (long long)B_ * H_ * W_) return;
  int b = (int)(i / (H_ * W_));
  atomicMax(&nmax[b], seg[i]);
}

__global__ void k_clamp(const int* __restrict__ lab, const int* __restrict__ nmax,
                        int* __restrict__ clustered, int* __restrict__ present) {
  int i = blockIdx.x * blockDim.x + threadIdx.x;
  if (i >= B_ * N_) return;
  int b = i / N_, n = i % N_;
  int c = lab[i];
  if (n >= nmax[b]) c = 0;
  clustered[i] = c;
  present[b * N_ + c] = 1;
}

__global__ void k_scan(int* present) {
  int b = threadIdx.x;
  if (b >= B_) return;
  int* row = present + b * N_;
  int s = 0;
  for (int i = 0; i < N_; ++i) {
    s += row[i];
    row[i] = s;
  }
}

__global__ void k_remap(const int* __restrict__ clustered, const int* __restrict__ scan,
                        int* __restrict__ c2) {
  int i = blockIdx.x * blockDim.x + threadIdx.x;
  if (i >= B_ * N_) return;
  int b = i / N_;
  c2[i] = scan[b * N_ + clustered[i]] - 1;
}

__global__ void k_segout(const int* __restrict__ seg, const int* __restrict__ c2,
                         int* __restrict__ oseg) {
  long long i = (long long)blockIdx.x * blockDim.x + threadIdx.x;
  if (i >= (long long)B_ * H_ * W_) return;
  int b = (int)(i / (H_ * W_));
  oseg[i] = c2[b * N_ + seg[i]];
}

// ---------------- x_agg = adj @ x : WMMA + async LDS staging ----------------
// Block = 256 threads (8 waves) -> 128(M) x 64(C) output tile.
// K loop in chunks of 32; adj tile (128x32, padded stride 36) and x slab (32x64)
// staged in LDS via GLOBAL_LOAD_ASYNC_TO_LDS_B128 (ASYNCcnt path).
#define KT_ 32
#define ASTR_ 36  // padded row stride (floats): 36*4 bytes = 16B-aligned rows, spreads banks

__global__ void __launch_bounds__(256) k_adjx(const float* __restrict__ adj,
                                              const float* __restrict__ x,
                                              float* __restrict__ y) {
  __shared__ float sA[128 * ASTR_];  // 18432 B
  __shared__ float sX[KT_ * 64];     // 8192 B
  const int CT = C_ / 64;    // 12
  const int MTB = N_ / 128;  // 16
  int blk = blockIdx.x;
  if (blk >= B_ * MTB * CT) return;
  int ct = blk % CT;
  int mtb = (blk / CT) % MTB;
  int b = blk / (CT * MTB);
  int tid = threadIdx.x;
  int wv = tid >> 5, lane = tid & 31;
  int half = lane >> 4, l16 = lane & 15;
  int m0 = mtb * 128, c0 = ct * 64;
  const float* Ag = adj + ((size_t)b * N_ + m0) * N_;
  const float* Xg = x + (size_t)b * N_ * C_;
  v8f acc0 = {}, acc1 = {}, acc2 = {}, acc3 = {};

  for (int k0 = 0; k0 < N_; k0 += KT_) {
    // ---- stage A tile: 128 rows x KT_ floats (1024 float4 chunks) ----
    for (int t = tid; t < 128 * (KT_ / 4); t += 256) {
      int r = t >> 3, q = (t & 7) * 4;  // row, float offset within chunk row
      const float* g = Ag + (size_t)r * N_ + k0 + q;
      float* l = sA + r * ASTR_ + q;
#if HAVE_ASYNC_LDS
      __builtin_amdgcn_global_load_async_to_lds_b128(GAS128(g), LAS128(l), 0, 0);
#else
      *(float4*)l = *(const float4*)g;
#endif
    }
    // ---- stage X slab: KT_ rows x 64 floats (512 float4 chunks) ----
    for (int t = tid; t < KT_ * 16; t += 256) {
      int r = t >> 4, q = (t & 15) * 4;
      const float* g = Xg + (size_t)(k0 + r) * C_ + c0 + q;
      float* l = sX + r * 64 + q;
#if HAVE_ASYNC_LDS
      __builtin_amdgcn_global_load_async_to_lds_b128(GAS128(g), LAS128(l), 0, 0);
#else
      *(float4*)l = *(const float4*)g;
#endif
    }
#if HAVE_ASYNC_LDS
    wait_async0();
#endif
    __syncthreads();

    const float* aRow = sA + (wv * 16 + l16) * ASTR_;
    for (int kk = 0; kk < KT_; kk += 4) {
      int kb = kk + half * 2;
      v2f a = {aRow[kb], aRow[kb + 1]};
      const float* x0 = sX + kb * 64 + l16;
      const float* x1 = x0 + 64;
      v2f b0 = {x0[0], x1[0]};
      v2f b1 = {x0[16], x1[16]};
      v2f b2 = {x0[32], x1[32]};
      v2f b3 = {x0[48], x1[48]};
      acc0 = __builtin_amdgcn_wmma_f32_16x16x4_f32(false, a, false, b0, (short)0, acc0, false, false);
      acc1 = __builtin_amdgcn_wmma_f32_16x16x4_f32(false, a, false, b1, (short)0, acc1, false, false);
      acc2 = __builtin_amdgcn_wmma_f32_16x16x4_f32(false, a, false, b2, (short)0, acc2, false, false);
      acc3 = __builtin_amdgcn_wmma_f32_16x16x4_f32(false, a, false, b3, (short)0, acc3, false, false);
    }
    __syncthreads();
  }

  for (int v = 0; v < 8; ++v) {
    int row = m0 + wv * 16 + v + 8 * half;
    float* o = y + ((size_t)b * N_ + row) * C_ + c0 + l16;
    o[0] = acc0[v];
    o[16] = acc1[v];
    o[32] = acc2[v];
    o[48] = acc3[v];
  }
}

// ---------------- segment mean ----------------
__global__ void k_segsum(const float* __restrict__ xagg, const int* __restrict__ c2,
                         float* __restrict__ out, float* __restrict__ cnt) {
  int bn = blockIdx.x;  // b*N + n
  if (bn >= B_ * N_) return;
  int b = bn / N_;
  int v = c2[bn];
  const float* src = xagg + (size_t)bn * C_;
  float* dst = out + ((size_t)b * N_ + v) * C_;
  for (int c = threadIdx.x; c < C_; c += blockDim.x) atomicAdd(&dst[c], src[c]);
  if (threadIdx.x == 0) atomicAdd(&cnt[b * N_ + v], 1.0f);
}

__global__ void k_segdiv(float* __restrict__ out, const float* __restrict__ cnt) {
  long long i = (long long)blockIdx.x * blockDim.x + threadIdx.x;
  if (i >= (long long)B_ * N_ * C_) return;
  int bn = (int)(i / C_);
  out[i] /= fmaxf(cnt[bn], 1.0f);
}

// ==================== launch ====================
extern "C" void kernel_launch(void* const* d_in, const int* in_sizes, int n_in,
                              void* d_out, int out_size, void* d_ws, size_t ws_size,
                              hipStream_t stream) {
  const float* x = (const float*)d_in[0];       // (B,N,C)
  const float* protos = (const float*)d_in[1];  // (K,C)
  const int* seg = (const int*)d_in[2];         // (B,1,H,W)

  float* out_x = (float*)d_out;                         // (B,N,C)
  int* out_seg = (int*)(out_x + (size_t)B_ * N_ * C_);  // (B,1,H,W) int32

  // ---- workspace layout ----
  char* ws = (char*)d_ws;
  size_t off = 0;
  auto alloc = [&](size_t bytes) {
    size_t o = off;
    off = (off + bytes + 255) & ~(size_t)255;
    return o;
  };
  float* adjP = (float*)(ws + alloc((size_t)B_ * N_ * N_ * 4));  // proximity counts
  float* adjW = (float*)(ws + alloc((size_t)B_ * N_ * N_ * 4));  // class adj -> fused adj
  float* xagg = (float*)(ws + alloc((size_t)B_ * N_ * C_ * 4));  // adj @ x
  float* cs = (float*)(ws + alloc((size_t)B_ * N_ * K_ * 4));
  float* pninv = (float*)(ws + alloc(K_ * 4));
  float* xninv = (float*)(ws + alloc((size_t)B_ * N_ * 4));
  float* dpinv = (float*)(ws + alloc((size_t)B_ * N_ * 4));
  float* dcinv = (float*)(ws + alloc((size_t)B_ * N_ * 4));
  int* labA = (int*)(ws + alloc((size_t)B_ * N_ * 4));
  int* labB = (int*)(ws + alloc((size_t)B_ * N_ * 4));
  int* clustered = (int*)(ws + alloc((size_t)B_ * N_ * 4));
  int* present = (int*)(ws + alloc((size_t)B_ * N_ * 4));
  int* c2 = (int*)(ws + alloc((size_t)B_ * N_ * 4));
  int* nmax = (int*)(ws + alloc(B_ * 4));
  float* cnt = (float*)(ws + alloc((size_t)B_ * N_ * 4));

  const int TB = 256;
  const long long NPIX = (long long)B_ * H_ * W_;

  // 1) cosine similarities
  k_pnorm<<<1, 16, 0, stream>>>(protos, pninv);
  k_xnorm<<<(B_ * N_ + TB - 1) / TB, TB, 0, stream>>>(x, xninv);
  k_cs<<<(B_ * N_ * K_ + TB - 1) / TB, TB, 0, stream>>>(x, protos, xninv, pninv, cs);

  // 2) class adjacency via WMMA (one wave per 16x16 tile)
  {
    int waves = B_ * (N_ / 16) * (N_ / 16);
    k_class_adj<<<(waves * 32 + TB - 1) / TB, TB, 0, stream>>>(cs, adjW);
  }

  // 3) proximity adjacency
  (void)hipMemsetAsync(adjP, 0, (size_t)B_ * N_ * N_ * 4, stream);
  k_prox<<<(int)((NPIX + TB - 1) / TB), TB, 0, stream>>>(seg, adjP);

  // 4) normalize both, fuse, set diag
  k_dinv<<<(B_ * N_ * 32 + TB - 1) / TB, TB, 0, stream>>>(adjP, dpinv);
  k_dinv<<<(B_ * N_ * 32 + TB - 1) / TB, TB, 0, stream>>>(adjW, dcinv);
  {
    long long tot = (long long)B_ * N_ * N_;
    k_fuse<<<(int)((tot + TB - 1) / TB), TB, 0, stream>>>(adjP, dpinv, dcinv, adjW);
  }

  // 5) connected components (24 fixed iterations, ping-pong; ends in labA)
  k_lab_init<<<(B_ * N_ + TB - 1) / TB, TB, 0, stream>>>(labA);
  for (int it = 0; it < 12; ++it) {
    k_cc<<<(B_ * N_ * 32 + TB - 1) / TB, TB, 0, stream>>>(adjW, labA, labB);
    k_cc<<<(B_ * N_ * 32 + TB - 1) / TB, TB, 0, stream>>>(adjW, labB, labA);
  }

  // 6) clamp by n_max, compact labels, remap segments
  (void)hipMemsetAsync(nmax, 0, B_ * 4, stream);
  k_nmax<<<(int)((NPIX + TB - 1) / TB), TB, 0, stream>>>(seg, nmax);
  (void)hipMemsetAsync(present, 0, (size_t)B_ * N_ * 4, stream);
  k_clamp<<<(B_ * N_ + TB - 1) / TB, TB, 0, stream>>>(labA, nmax, clustered, present);
  k_scan<<<1, 32, 0, stream>>>(present);
  k_remap<<<(B_ * N_ + TB - 1) / TB, TB, 0, stream>>>(clustered, present, c2);
  k_segout<<<(int)((NPIX + TB - 1) / TB), TB, 0, stream>>>(seg, c2, out_seg);

  // 7) x = adj @ x via WMMA with async-LDS staged tiles
  {
    int blocks = B_ * (N_ / 128) * (C_ / 64);  // 768
    k_adjx<<<blocks, 256, 0, stream>>>(adjW, x, xagg);
  }

  // 8) segment mean into output
  (void)hipMemsetAsync(out_x, 0, (size_t)B_ * N_ * C_ * 4, stream);
  (void)hipMemsetAsync(cnt, 0, (size_t)B_ * N_ * 4, stream);
  k_segsum<<<B_ * N_, TB, 0, stream>>>(xagg, c2, out_x, cnt);
  {
    long long tot = (long long)B_ * N_ * C_;
    k_segdiv<<<(int)((tot + TB - 1) / TB), TB, 0, stream>>>(out_x, cnt);
  }
  (void)in_sizes; (void)n_in; (void)out_size; (void)ws_size;
}